// GAT_47579647705320
// MI455X (gfx1250) — compile-verified
//
#include <hip/hip_runtime.h>
#include <math.h>

typedef float v2f __attribute__((ext_vector_type(2)));
typedef float v8f __attribute__((ext_vector_type(8)));

#define NEG_SLOPE 0.2f
#define BN_EPS 1e-5f

#define WMMA_F32(a, b, c) \
  __builtin_amdgcn_wmma_f32_16x16x4_f32(false, (a), false, (b), (short)0, (c), false, false)

// ---------------------------------------------------------------------------
// Dense GEMM: C[M,Np] = A[M,K] * B[K,Np], row-major f32, Np % 64 == 0.
// V_WMMA_F32_16X16X4_F32. One wave -> 32x64 tile: two M tiles share the B
// fragments (8 B loads feed 8 WMMAs per k-step). 4 waves/block -> 128 rows.
// No bounds checks: M % 32 == 0, K % 4 == 0, Np padded (pad_w_kernel).
// 32-bit element offsets from uniform bases -> SADDR-form global loads.
// ---------------------------------------------------------------------------
__global__ __launch_bounds__(128)
void gemm_wmma_f32(const float* __restrict__ A, const float* __restrict__ B,
                   float* __restrict__ C, int M, int K, int Np) {
  const int lane = threadIdx.x & 31;
  const int wave = threadIdx.x >> 5;
  const int m0 = (blockIdx.x * 4 + wave) * 32;   // 32 rows per wave
  if (m0 >= M) return;
  const int n0 = blockIdx.y * 64;

  const int row  = lane & 15;     // A row / B,C column within tile
  const int lhi  = lane >> 4;     // 0: K pair {0,1}, C rows 0..7 ; 1: {2,3}, 8..15
  const int koff = lhi * 2;

  unsigned aoff0 = (unsigned)(m0 + row) * (unsigned)K + (unsigned)koff;
  unsigned aoff1 = aoff0 + 16u * (unsigned)K;
  unsigned boff  = (unsigned)koff * (unsigned)Np + (unsigned)(n0 + row);

  v8f c00 = {}; v8f c01 = {}; v8f c02 = {}; v8f c03 = {};
  v8f c10 = {}; v8f c11 = {}; v8f c12 = {}; v8f c13 = {};

#pragma unroll 2
  for (int k = 0; k < K; k += 4) {
    // A fragments: lanes 0-15 hold K=k,k+1 ; lanes 16-31 hold K=k+2,k+3
    v2f a0 = *(const v2f*)(A + aoff0);
    v2f a1 = *(const v2f*)(A + aoff1);
    const float* __restrict__ Bp = B + boff;     // row k+koff, col n0+row
    v2f b0, b1, b2, b3;
    b0.x = Bp[0];   b0.y = Bp[Np];
    b1.x = Bp[16];  b1.y = Bp[Np + 16];
    b2.x = Bp[32];  b2.y = Bp[Np + 32];
    b3.x = Bp[48];  b3.y = Bp[Np + 48];
    c00 = WMMA_F32(a0, b0, c00);
    c01 = WMMA_F32(a0, b1, c01);
    c02 = WMMA_F32(a0, b2, c02);
    c03 = WMMA_F32(a0, b3, c03);
    c10 = WMMA_F32(a1, b0, c10);
    c11 = WMMA_F32(a1, b1, c11);
    c12 = WMMA_F32(a1, b2, c12);
    c13 = WMMA_F32(a1, b3, c13);
    aoff0 += 4u; aoff1 += 4u; boff += 4u * (unsigned)Np;
  }

  // C/D layout: VGPR v -> M = v + 8*lhi, N = lane&15
  unsigned coff0 = (unsigned)(m0 + lhi * 8) * (unsigned)Np + (unsigned)(n0 + row);
  unsigned coff1 = coff0 + 16u * (unsigned)Np;
#pragma unroll
  for (int v = 0; v < 8; ++v) {
    C[coff0]      = c00[v];
    C[coff0 + 16] = c01[v];
    C[coff0 + 32] = c02[v];
    C[coff0 + 48] = c03[v];
    C[coff1]      = c10[v];
    C[coff1 + 16] = c11[v];
    C[coff1 + 32] = c12[v];
    C[coff1 + 48] = c13[v];
    coff0 += (unsigned)Np; coff1 += (unsigned)Np;
  }
}

// zero-pad W[K,F] -> Wp[K,Fp]
__global__ void pad_w_kernel(const float* __restrict__ W, float* __restrict__ Wp,
                             int K, int F, int Fp) {
  int i = blockIdx.x * blockDim.x + threadIdx.x;
  if (i >= K * Fp) return;
  int k = i / Fp, j = i - k * Fp;
  Wp[i] = (j < F) ? W[(size_t)k * F + j] : 0.f;
}

// ---------------------------------------------------------------------------
// alpha_s[n,h] = sum_c lin[n, h*C+c] * a_src[h,c] ; alpha_d likewise.
// lin has row stride Fs (padded), logical width H*C.
// ---------------------------------------------------------------------------
__global__ void alpha_kernel(const float* __restrict__ lin,
                             const float* __restrict__ a_s, const float* __restrict__ a_d,
                             float* __restrict__ out_s, float* __restrict__ out_d,
                             int NH, int H, int Cc, int Fs) {
  int i = blockIdx.x * blockDim.x + threadIdx.x;
  if (i >= NH) return;
  int n = i / H, h = i - n * H;
  const float* hp  = lin + (size_t)n * Fs + h * Cc;
  const float* asp = a_s + h * Cc;
  const float* adp = a_d + h * Cc;
  float ss = 0.f, sd = 0.f;
  for (int c = 0; c < Cc; ++c) { float v = hp[c]; ss += v * asp[c]; sd += v * adp[c]; }
  out_s[i] = ss; out_d[i] = sd;
}

__global__ void fill_kernel(float* __restrict__ p, float val, int n) {
  int i = blockIdx.x * blockDim.x + threadIdx.x;
  if (i < n) p[i] = val;
}

__device__ __forceinline__ void atomicMaxF(float* addr, float val) {
  if (val >= 0.f) atomicMax((int*)addr, __float_as_int(val));
  else            atomicMin((unsigned int*)addr, (unsigned int)__float_as_int(val));
}

__device__ __forceinline__ float leaky(float x) {
  return x > 0.f ? x : NEG_SLOPE * x;
}

// segment max of leaky_relu(alpha_s[src]+alpha_d[dst]) grouped by dst
__global__ void edge_max_kernel(const int* __restrict__ ei, int E,
                                const float* __restrict__ as_b, const float* __restrict__ ad_b,
                                float* __restrict__ m_b, int H, int EH) {
  int i = blockIdx.x * blockDim.x + threadIdx.x;
  if (i >= EH) return;
  int e = i / H, h = i - e * H;
  int src = (e < E) ? ei[e]     : (e - E);
  int dst = (e < E) ? ei[E + e] : (e - E);
  float ev = leaky(as_b[(size_t)src * H + h] + ad_b[(size_t)dst * H + h]);
  atomicMaxF(&m_b[(size_t)dst * H + h], ev);
}

// p = exp(e - m[dst]); denom[dst] += p; p saved for the scatter pass
__global__ void edge_exp_kernel(const int* __restrict__ ei, int E,
                                const float* __restrict__ as_b, const float* __restrict__ ad_b,
                                const float* __restrict__ m_b, float* __restrict__ d_b,
                                float* __restrict__ p_b, int H, int EH) {
  int i = blockIdx.x * blockDim.x + threadIdx.x;
  if (i >= EH) return;
  int e = i / H, h = i - e * H;
  int src = (e < E) ? ei[e]     : (e - E);
  int dst = (e < E) ? ei[E + e] : (e - E);
  float ev = leaky(as_b[(size_t)src * H + h] + ad_b[(size_t)dst * H + h]);
  float mm = m_b[(size_t)dst * H + h];
  if (mm == -INFINITY) mm = 0.f;          // reference: where(isfinite(m), m, 0)
  float p = __expf(ev - mm);
  p_b[i] = p;
  atomicAdd(&d_b[(size_t)dst * H + h], p);
}

// acc[dst, h*C+c] += (p/denom) * lin[src, h*C+c]  (one block per edge, coalesced)
// lin row stride Fs (padded); acc row stride F (logical).
__global__ void scatter_kernel(const int* __restrict__ ei, int E,
                               const float* __restrict__ lin, const float* __restrict__ p_b,
                               const float* __restrict__ d_b, float* __restrict__ acc,
                               int H, int Cc, int F, int Fs) {
  int e = blockIdx.x;
  int src = (e < E) ? ei[e]     : (e - E);
  int dst = (e < E) ? ei[E + e] : (e - E);
  __shared__ float coef[8];
  if ((int)threadIdx.x < H)
    coef[threadIdx.x] = p_b[(size_t)e * H + threadIdx.x] /
                        (d_b[(size_t)dst * H + threadIdx.x] + 1e-16f);
  __syncthreads();
  const float* __restrict__ s = lin + (size_t)src * Fs;
  float* __restrict__ d = acc + (size_t)dst * F;
  for (int f = threadIdx.x; f < F; f += blockDim.x)
    atomicAdd(&d[f], coef[f / Cc] * s[f]);
}

// in-place: x = prelu(bn(x + bias))
__global__ void post_kernel(float* __restrict__ x, const float* __restrict__ b,
                            const float* __restrict__ g, const float* __restrict__ be,
                            const float* __restrict__ mm, const float* __restrict__ vv,
                            const float* __restrict__ pw, int total, int F) {
  int i = blockIdx.x * blockDim.x + threadIdx.x;
  if (i >= total) return;
  int f = i % F;
  float v = x[i] + b[f];
  v = (v - mm[f]) * (g[f] * rsqrtf(vv[f] + BN_EPS)) + be[f];
  float w = pw[0];
  x[i] = v > 0.f ? v : w * v;
}

// last layer: mean over 5 heads of 40 + bias, then log_softmax over 40
__global__ void final_kernel(const float* __restrict__ acc, const float* __restrict__ b4,
                             float* __restrict__ out, int Nn) {
  int n = blockIdx.x * blockDim.x + threadIdx.x;
  if (n >= Nn) return;
  const float* a = acc + (size_t)n * 200;
  float v[40];
  float mx = -INFINITY;
#pragma unroll
  for (int c = 0; c < 40; ++c) {
    float s = 0.f;
#pragma unroll
    for (int h = 0; h < 5; ++h) s += a[h * 40 + c];
    v[c] = s * 0.2f + b4[c];
    mx = fmaxf(mx, v[c]);
  }
  float se = 0.f;
#pragma unroll
  for (int c = 0; c < 40; ++c) se += __expf(v[c] - mx);
  float ls = __logf(se) + mx;
  float* o = out + (size_t)n * 40;
#pragma unroll
  for (int c = 0; c < 40; ++c) o[c] = v[c] - ls;
}

extern "C" void kernel_launch(void* const* d_in, const int* in_sizes, int n_in,
                              void* d_out, int out_size, void* d_ws, size_t ws_size,
                              hipStream_t stream) {
  const float* x  = (const float*)d_in[0];
  const int*   ei = (const int*)d_in[1];
  const int Nn = in_sizes[0] / 128;     // 20000
  const int Ee = in_sizes[1] / 2;       // 320000
  const int EP = Ee + Nn;               // + self loops

  // workspace layout (floats)
  float* ws    = (float*)d_ws;
  float* S_lin = ws;                               // N*512 : GEMM output h (padded stride)
  float* S_act = S_lin + (size_t)Nn * 512;         // N*512 : acc / activation
  float* as_b  = S_act + (size_t)Nn * 512;         // N*8
  float* ad_b  = as_b + (size_t)Nn * 8;            // N*8
  float* m_b   = ad_b + (size_t)Nn * 8;            // N*8
  float* d_b   = m_b  + (size_t)Nn * 8;            // N*8
  float* p_b   = d_b  + (size_t)Nn * 8;            // EP*8
  float* W_pad = p_b  + (size_t)EP * 8;            // 448*256 (layer-4 padded W)

  const int Kd[4] = {128, 512, 448, 448};
  const int Fd[4] = {512, 448, 448, 200};   // logical output width
  const int Fp[4] = {512, 448, 448, 256};   // padded to multiple of 64
  const int Hs[4] = {8, 7, 7, 5};
  const int Cs[4] = {64, 64, 64, 40};
  const int wI[4] = {2, 10, 18, 26};
  const float* pw = (const float*)d_in[30];

  const float* act_in = x;
  for (int l = 0; l < 4; ++l) {
    const int K = Kd[l], F = Fd[l], Fs = Fp[l], H = Hs[l], Cc = Cs[l];
    const float* W   = (const float*)d_in[wI[l] + 0];
    const float* a_s = (const float*)d_in[wI[l] + 1];
    const float* a_d = (const float*)d_in[wI[l] + 2];
    const float* b   = (const float*)d_in[wI[l] + 3];

    const float* Wuse = W;
    if (Fs != F) {
      pad_w_kernel<<<(K * Fs + 255) / 256, 256, 0, stream>>>(W, W_pad, K, F, Fs);
      Wuse = W_pad;
    }

    dim3 gg((Nn + 127) / 128, Fs / 64);
    gemm_wmma_f32<<<gg, 128, 0, stream>>>(act_in, Wuse, S_lin, Nn, K, Fs);

    const int NH = Nn * H;
    alpha_kernel<<<(NH + 255) / 256, 256, 0, stream>>>(S_lin, a_s, a_d, as_b, ad_b, NH, H, Cc, Fs);
    fill_kernel<<<(NH + 255) / 256, 256, 0, stream>>>(m_b, -INFINITY, NH);
    fill_kernel<<<(NH + 255) / 256, 256, 0, stream>>>(d_b, 0.f, NH);
    fill_kernel<<<((Nn * F) + 255) / 256, 256, 0, stream>>>(S_act, 0.f, Nn * F);

    const int EH = EP * H;
    edge_max_kernel<<<(EH + 255) / 256, 256, 0, stream>>>(ei, Ee, as_b, ad_b, m_b, H, EH);
    edge_exp_kernel<<<(EH + 255) / 256, 256, 0, stream>>>(ei, Ee, as_b, ad_b, m_b, d_b, p_b, H, EH);
    scatter_kernel<<<EP, 256, 0, stream>>>(ei, Ee, S_lin, p_b, d_b, S_act, H, Cc, F, Fs);

    if (l < 3) {
      const float* g  = (const float*)d_in[wI[l] + 4];
      const float* be = (const float*)d_in[wI[l] + 5];
      const float* mm = (const float*)d_in[wI[l] + 6];
      const float* vv = (const float*)d_in[wI[l] + 7];
      post_kernel<<<((Nn * F) + 255) / 256, 256, 0, stream>>>(S_act, b, g, be, mm, vv, pw, Nn * F, F);
      act_in = S_act;
    } else {
      final_kernel<<<(Nn + 127) / 128, 128, 0, stream>>>(S_act, b, (float*)d_out, Nn);
    }
  }
}